// GCL_40836549050565
// MI455X (gfx1250) — compile-verified
//
#include <hip/hip_runtime.h>

typedef __attribute__((ext_vector_type(2))) float v2f;
typedef __attribute__((ext_vector_type(8))) float v8f;

#define NNODES 50000
#define NEDGES 600000

// ---------------------------------------------------------------------------
// Dense GEMM  H[M,N] = X[M,K] @ W[K,N]   (fp32, V_WMMA_F32_16X16X4_F32)
//
// One wave computes a 16-row stripe across ALL N columns (NT = N/16 tiles of
// 16x16 kept in registers), looping K in steps of 4.  X is therefore read
// exactly once from memory; W is tiny and stays L2/L0 resident.
//
// ISA layouts (CDNA5 ISA):
//   A 16x4 f32 : lane&15 = M row, VGPR0/1 = k, lane[4] selects k-half (+2)
//   B  4x16 f32: lane&15 = N col, VGPR0/1 = k, lane[4] selects k-half (+2)
//   C/D 16x16  : VGPR j -> M = j (lanes 0-15) / j+8 (lanes 16-31), lane&15 = N
// ---------------------------------------------------------------------------
template <int K, int N>
__global__ __launch_bounds__(256) void gemm_wmma_f32(
    const float* __restrict__ X, const float* __restrict__ W,
    float* __restrict__ H, int M) {
  constexpr int NT = N / 16;
  const int wave = (blockIdx.x * blockDim.x + threadIdx.x) >> 5;
  const int lane = threadIdx.x & 31;
  const int row0 = wave * 16;
  if (row0 >= M) return;  // uniform per-wave; active waves keep EXEC all-ones

  const int m  = lane & 15;        // A: row within tile.  B/D: col within tile
  const int kh = (lane >> 4) * 2;  // k-half offset inside the 4-wide k step

  v8f acc[NT];
#pragma unroll
  for (int t = 0; t < NT; ++t) {
    v8f z = {};
    acc[t] = z;
  }

  const float* __restrict__ xrow = X + (size_t)(row0 + m) * K;

  for (int k0 = 0; k0 < K; k0 += 4) {
    v2f a;  // contiguous pair -> single global_load_b64 per lane
    a.x = xrow[k0 + kh + 0];
    a.y = xrow[k0 + kh + 1];
#pragma unroll
    for (int t = 0; t < NT; ++t) {
      v2f b;  // lanes 0-15 read 64B contiguous per k row (coalesced)
      b.x = W[(size_t)(k0 + kh + 0) * N + t * 16 + m];
      b.y = W[(size_t)(k0 + kh + 1) * N + t * 16 + m];
      acc[t] = __builtin_amdgcn_wmma_f32_16x16x4_f32(
          /*neg_a=*/false, a, /*neg_b=*/false, b,
          /*c_mod=*/(short)0, acc[t], /*reuse_a=*/false, /*reuse_b=*/false);
    }
  }

  const int rbase = row0 + ((lane >> 4) << 3);  // lanes 16-31 hold rows +8
#pragma unroll
  for (int t = 0; t < NT; ++t) {
#pragma unroll
    for (int j = 0; j < 8; ++j) {
      H[(size_t)(rbase + j) * N + t * 16 + m] = acc[t][j];
    }
  }
}

// ---------------------------------------------------------------------------
// acc[i, f] = bias[f]   (seeds the segment-sum accumulator with +b for free)
// ---------------------------------------------------------------------------
template <int F>
__global__ void init_bias(float* __restrict__ acc, const float* __restrict__ b,
                          int total) {
  int i = blockIdx.x * blockDim.x + threadIdx.x;
  if (i < total) acc[i] = b[i & (F - 1)];
}

// ---------------------------------------------------------------------------
// Edge scatter:  acc[dst[e], :] += vals[e] * h[src[e], :]
// One thread handles one float4 chunk; F/4 consecutive threads cover an edge,
// so each source row is gathered as one coalesced 256/512B burst, and the
// reduction uses global_atomic_add_f32 (L2-side atomics).
// ---------------------------------------------------------------------------
template <int F>
__global__ void scatter_edges(float* __restrict__ acc,
                              const float* __restrict__ h,
                              const int* __restrict__ src,
                              const int* __restrict__ dst,
                              const float* __restrict__ vals) {
  constexpr int C = F / 4;
  int idx = blockIdx.x * blockDim.x + threadIdx.x;
  if (idx >= NEDGES * C) return;
  const int e = idx / C;
  const int c = idx - e * C;
  const int s = src[e];
  const int d = dst[e];
  const float v = vals[e];
  const float4 x = ((const float4*)(h + (size_t)s * F))[c];
  float* out = acc + (size_t)d * F + c * 4;
  atomicAdd(out + 0, v * x.x);
  atomicAdd(out + 1, v * x.y);
  atomicAdd(out + 2, v * x.z);
  atomicAdd(out + 3, v * x.w);
}

// ---------------------------------------------------------------------------
__global__ void relu_inplace(float* __restrict__ x, int n) {
  int i = blockIdx.x * blockDim.x + threadIdx.x;
  if (i < n) x[i] = fmaxf(x[i], 0.0f);
}

// ---------------------------------------------------------------------------
static inline int cdiv(long long a, int b) { return (int)((a + b - 1) / b); }

static void run_gcn(const float* x, const int* src, const int* dst,
                    const float* vals, const float* W0, const float* b0,
                    const float* W1, const float* b1, float* out,
                    float* ws_xw, float* ws_acc, float* ws_h2,
                    hipStream_t stream) {
  constexpr int M = NNODES;
  const int gemm_waves = M / 16;  // 3125 (M divisible by 16)
  const int gemm_blocks = cdiv((long long)gemm_waves * 32, 256);

  // ---- layer 1: h = relu(spmm(vals, x @ W0) + b0) ----
  gemm_wmma_f32<128, 128><<<gemm_blocks, 256, 0, stream>>>(x, W0, ws_xw, M);
  init_bias<128><<<cdiv(M * 128, 256), 256, 0, stream>>>(ws_acc, b0, M * 128);
  scatter_edges<128><<<cdiv((long long)NEDGES * 32, 256), 256, 0, stream>>>(
      ws_acc, ws_xw, src, dst, vals);
  relu_inplace<<<cdiv(M * 128, 256), 256, 0, stream>>>(ws_acc, M * 128);

  // ---- layer 2: out = spmm(vals, h @ W1) + b1 ----
  gemm_wmma_f32<128, 64><<<gemm_blocks, 256, 0, stream>>>(ws_acc, W1, ws_h2, M);
  init_bias<64><<<cdiv(M * 64, 256), 256, 0, stream>>>(out, b1, M * 64);
  scatter_edges<64><<<cdiv((long long)NEDGES * 16, 256), 256, 0, stream>>>(
      out, ws_h2, src, dst, vals);
}

extern "C" void kernel_launch(void* const* d_in, const int* in_sizes, int n_in,
                              void* d_out, int out_size, void* d_ws,
                              size_t ws_size, hipStream_t stream) {
  (void)in_sizes; (void)n_in; (void)out_size; (void)ws_size;

  const float* x1    = (const float*)d_in[0];
  const int*   src1  = (const int*)d_in[1];
  const int*   dst1  = (const int*)d_in[2];
  const float* vals1 = (const float*)d_in[3];
  const float* x2    = (const float*)d_in[4];
  const int*   src2  = (const int*)d_in[5];
  const int*   dst2  = (const int*)d_in[6];
  const float* vals2 = (const float*)d_in[7];
  const float* W1_0  = (const float*)d_in[8];
  const float* b1_0  = (const float*)d_in[9];
  const float* W1_1  = (const float*)d_in[10];
  const float* b1_1  = (const float*)d_in[11];
  const float* W2_0  = (const float*)d_in[12];
  const float* b2_0  = (const float*)d_in[13];
  const float* W2_1  = (const float*)d_in[14];
  const float* b2_1  = (const float*)d_in[15];

  float* out = (float*)d_out;  // [2, NNODES, 64]

  // workspace partition (reused across both graphs via stream ordering)
  float* ws_xw  = (float*)d_ws;                       // [NNODES,128]
  float* ws_acc = ws_xw + (size_t)NNODES * 128;       // [NNODES,128]
  float* ws_h2  = ws_acc + (size_t)NNODES * 128;      // [NNODES, 64]

  run_gcn(x1, src1, dst1, vals1, W1_0, b1_0, W1_1, b1_1,
          out, ws_xw, ws_acc, ws_h2, stream);
  run_gcn(x2, src2, dst2, vals2, W2_0, b2_0, W2_1, b2_1,
          out + (size_t)NNODES * 64, ws_xw, ws_acc, ws_h2, stream);
}